// DGCNN_ReID_60189671686683
// MI455X (gfx1250) — compile-verified
//
#include <hip/hip_runtime.h>
#include <math.h>
#include <stdint.h>

typedef __attribute__((ext_vector_type(2))) float v2f;
typedef __attribute__((ext_vector_type(8))) float v8f;

#define KNN_K 20
#define SLOPE 0.2f
#define EPS_BN 1e-5f
#define NEG_BIG -1e30f

#if defined(__has_builtin)
#if __has_builtin(__builtin_amdgcn_global_load_async_to_lds_b128) && \
    __has_builtin(__builtin_amdgcn_s_wait_asynccnt)
#define USE_ASYNC_LDS 1
#endif
#endif
#ifndef USE_ASYNC_LDS
#define USE_ASYNC_LDS 0
#endif

#if USE_ASYNC_LDS
typedef int avi4 __attribute__((vector_size(4 * sizeof(int))));
typedef __attribute__((address_space(1))) avi4 g_avi4;
typedef __attribute__((address_space(3))) avi4 l_avi4;
#endif

__device__ inline v8f wmma4(v2f a, v2f b, v8f c) {
  // V_WMMA_F32_16X16X4_F32: D = A(16x4) * B(4x16) + C(16x16)
  return __builtin_amdgcn_wmma_f32_16x16x4_f32(false, a, false, b, (short)0, c,
                                               false, false);
}

__device__ inline float lrelu(float v) { return v > 0.f ? v : SLOPE * v; }
__device__ inline v2f ldv2(const float* p) { return *(const v2f*)p; }

// ------------------------------------------------- pad pts (B*N,3)->(B*N,4)
__global__ void pad_pts_kernel(const float* __restrict__ pts,
                               float* __restrict__ pts4, int total) {
  int r = blockIdx.x * blockDim.x + threadIdx.x;
  if (r >= total) return;
  pts4[(size_t)r * 4 + 0] = pts[(size_t)r * 3 + 0];
  pts4[(size_t)r * 4 + 1] = pts[(size_t)r * 3 + 1];
  pts4[(size_t)r * 4 + 2] = pts[(size_t)r * 3 + 2];
  pts4[(size_t)r * 4 + 3] = 0.f;
}

// ------------------------- repack w1 (64x6) -> (64x8) for padded features
// padded edge feature = [d0,d1,d2,0, c0,c1,c2,0]
__global__ void pad_w1_kernel(const float* __restrict__ w1,
                              float* __restrict__ w1p) {
  int o = blockIdx.x * blockDim.x + threadIdx.x;
  if (o >= 64) return;
  w1p[o * 8 + 0] = w1[o * 6 + 0];
  w1p[o * 8 + 1] = w1[o * 6 + 1];
  w1p[o * 8 + 2] = w1[o * 6 + 2];
  w1p[o * 8 + 3] = 0.f;
  w1p[o * 8 + 4] = w1[o * 6 + 3];
  w1p[o * 8 + 5] = w1[o * 6 + 4];
  w1p[o * 8 + 6] = w1[o * 6 + 5];
  w1p[o * 8 + 7] = 0.f;
}

// --------------- fold the (nbr - ctr) into weights: W' = [Wd, Wc - Wd]
// so y = W @ [nbr-ctr, ctr] == W' @ [nbr, ctr]  (raw gather, async-friendly)
__global__ void repack_w_kernel(const float* __restrict__ W,
                                float* __restrict__ Wm, int C, int O) {
  int t = blockIdx.x * blockDim.x + threadIdx.x;
  if (t >= O * C) return;
  int o = t / C, c = t - o * C;
  float wd = W[(size_t)o * 2 * C + c];
  float wc = W[(size_t)o * 2 * C + C + c];
  Wm[(size_t)o * 2 * C + c] = wd;
  Wm[(size_t)o * 2 * C + C + c] = wc - wd;
}

// ---------------------------------------------------------------- rownorm
__global__ void rownorm_kernel(const float* __restrict__ X, int stride, int C,
                               float* __restrict__ xx, int total) {
  int r = blockIdx.x * blockDim.x + threadIdx.x;
  if (r >= total) return;
  float s = 0.f;
  for (int c = 0; c < C; c++) {
    float v = X[(size_t)r * stride + c];
    s += v * v;
  }
  xx[r] = s;
}

// ------------------------------------------------- neg-dist via WMMA Gram
// nd[b,i,j] = 2 * <x_i, x_j> - |x_i|^2 - |x_j|^2     (C multiple of 4)
__global__ void negdist_kernel(const float* __restrict__ X, int stride, int C,
                               const float* __restrict__ xx,
                               float* __restrict__ nd, int B, int N) {
  int NT = N >> 4;
  int wid = blockIdx.x * (blockDim.x >> 5) + (threadIdx.x >> 5);
  int tilesPerB = NT * NT;
  if (wid >= B * tilesPerB) return;
  int b = wid / tilesPerB, rem = wid % tilesPerB;
  int it = rem / NT, jt = rem % NT;
  int lane = threadIdx.x & 31;
  int ln = lane & 15, hk = (lane >> 4) << 1;
  size_t irow = (size_t)b * N + it * 16 + ln;
  size_t jrow = (size_t)b * N + jt * 16 + ln;
  const float* Xi = X + irow * stride + hk;
  const float* Xj = X + jrow * stride + hk;
  v8f acc = {};
  int ksteps = C >> 2;
#pragma unroll 4
  for (int ks = 0; ks < ksteps; ks++) {
    acc = wmma4(ldv2(Xi + ks * 4), ldv2(Xj + ks * 4), acc);
  }
  int mo = (lane >> 4) * 8;
  int j = jt * 16 + ln;
  float xxj = xx[(size_t)b * N + j];
#pragma unroll
  for (int r = 0; r < 8; r++) {
    int i = it * 16 + mo + r;
    nd[((size_t)b * N + i) * N + j] = 2.f * acc[r] - xx[(size_t)b * N + i] - xxj;
  }
}

// ----------------------------------------------------------- top-k (k=20)
__global__ void topk_kernel(const float* __restrict__ nd, int* __restrict__ idx,
                            int BN, int N) {
  int r = blockIdx.x * blockDim.x + threadIdx.x;
  if (r >= BN) return;
  const float* row = nd + (size_t)r * N;
  float bv[KNN_K];
  int bi[KNN_K];
  for (int i = 0; i < KNN_K; i++) { bv[i] = NEG_BIG; bi[i] = 0; }
  for (int j = 0; j < N; j++) {
    float v = row[j];
    if (v > bv[KNN_K - 1]) {
      int p = KNN_K - 1;
      while (p > 0 && bv[p - 1] < v) {
        bv[p] = bv[p - 1];
        bi[p] = bi[p - 1];
        p--;
      }
      bv[p] = v;
      bi[p] = j;
    }
  }
  for (int i = 0; i < KNN_K; i++) idx[(size_t)r * KNN_K + i] = bi[i];
}

// --------------------------------------------------------------- zeroing
__global__ void zero_kernel(float* p, int n) {
  int t = blockIdx.x * blockDim.x + threadIdx.x;
  if (t < n) p[t] = 0.f;
}

// -------------------------------------- edge conv (two passes, WMMA core)
// Edge tile (raw gather, subtraction folded into repacked weights):
//   se[k][0:C]   = x[b, knn[b,n,k], :]
//   se[k][C:2C]  = x[b, n, :]
// pass 1: per-channel sum / sumsq of y = se @ W'^T over (B,N,K)
// pass 2: y*scale+shift -> leaky relu -> max over k -> Xout[b,n,o]
// Requires: C multiple of 4 (so 2C too), O multiple of 16.
__global__ void edge_pass_kernel(const float* __restrict__ X, int stride,
                                 const int* __restrict__ knn,
                                 const float* __restrict__ W, int C, int O,
                                 float* __restrict__ stats,
                                 const float* __restrict__ scsh,
                                 float* __restrict__ Xout, int ostride,
                                 int pass, int N) {
  int bn = blockIdx.x;  // b*N + n
  extern __shared__ char smemraw[];
  int* sidx = (int*)smemraw;
  float* se = (float*)(smemraw + 128);
  int twoC = 2 * C;
  if (threadIdx.x < KNN_K)
    sidx[threadIdx.x] = knn[(size_t)bn * KNN_K + threadIdx.x];
  __syncthreads();

  int lane = threadIdx.x & 31;
  int wave = threadIdx.x >> 5;
  int nwave = blockDim.x >> 5;
  int ln = lane & 15, hk = (lane >> 4) << 1, mo = (lane >> 4) * 8;
  int OT = O >> 4;
  int b = bn / N;
  int ksteps = twoC >> 2;
  int q4 = twoC >> 2;  // 16B quads per LDS row

  float s0[2] = {0.f, 0.f}, s1[2] = {0.f, 0.f};
  float rmx[2] = {NEG_BIG, NEG_BIG};

  for (int mtile = 0; mtile < 2; mtile++) {
    // stage 16 edge rows (k = mtile*16 + r) of width 2C into LDS, 16B/lane
    for (int e = threadIdx.x; e < 16 * q4; e += blockDim.x) {
      int r = e / q4, cq = e - r * q4;
      int c = cq * 4;
      int k = mtile * 16 + r;
      float* dst = se + r * twoC + c;
      if (k < KNN_K) {
        int inD = c < C;
        size_t srow = inD ? ((size_t)b * N + sidx[k]) : (size_t)bn;
        int cc = inD ? c : c - C;
        const float* src = X + srow * stride + cc;
#if USE_ASYNC_LDS
        __builtin_amdgcn_global_load_async_to_lds_b128(
            (g_avi4*)(unsigned long long)(uintptr_t)src, (l_avi4*)dst, 0, 0);
#else
        *(float4*)dst = *(const float4*)src;
#endif
      } else {
        float4 z = {0.f, 0.f, 0.f, 0.f};
        *(float4*)dst = z;
      }
    }
#if USE_ASYNC_LDS
    __builtin_amdgcn_s_wait_asynccnt(0);
#endif
    __syncthreads();

    int ci = 0;
    for (int ct = wave; ct < OT && ci < 2; ct += nwave, ci++) {
      int o = ct * 16 + ln;
      const float* serow = se + ln * twoC + hk;
      const float* wrow = W + (size_t)o * twoC + hk;
      v8f acc = {};
#pragma unroll 4
      for (int ks = 0; ks < ksteps; ks++) {
        acc = wmma4(ldv2(serow + ks * 4), ldv2(wrow + ks * 4), acc);
      }
      if (pass == 1) {
#pragma unroll
        for (int r = 0; r < 8; r++) {
          int k = mtile * 16 + mo + r;
          if (k < KNN_K) {
            float v = acc[r];
            s0[ci] += v;
            s1[ci] += v * v;
          }
        }
      } else {
        float sc = scsh[o], sh = scsh[O + o];
#pragma unroll
        for (int r = 0; r < 8; r++) {
          int k = mtile * 16 + mo + r;
          if (k < KNN_K) {
            float v = lrelu(acc[r] * sc + sh);
            rmx[ci] = fmaxf(rmx[ci], v);
          }
        }
      }
    }
    __syncthreads();
  }

  int ci = 0;
  for (int ct = wave; ct < OT && ci < 2; ct += nwave, ci++) {
    int o = ct * 16 + ln;
    if (pass == 1) {
      float a = s0[ci] + __shfl_xor(s0[ci], 16, 32);
      float q = s1[ci] + __shfl_xor(s1[ci], 16, 32);
      if (lane < 16) {
        atomicAdd(&stats[o], a);
        atomicAdd(&stats[O + o], q);
      }
    } else {
      float m = fmaxf(rmx[ci], __shfl_xor(rmx[ci], 16, 32));
      if (lane < 16) Xout[(size_t)bn * ostride + o] = m;
    }
  }
}

// ---------------------------------------------- generic WMMA GEMM: A @ W^T
// A: M x Kd (row stride lda), W: Nd x Kd row-major, Out: M x Nd (stride ldo)
// Requires: M multiple of 16, Kd multiple of 4. Nd may be ragged (store-guarded).
__global__ void gemm_wt_kernel(const float* __restrict__ A, int lda,
                               const float* __restrict__ W,
                               const float* __restrict__ bias,
                               float* __restrict__ Out, int ldo, int M, int Kd,
                               int Nd, float* __restrict__ stats) {
  int MT = M >> 4, NTt = (Nd + 15) >> 4;
  int wid = blockIdx.x * (blockDim.x >> 5) + (threadIdx.x >> 5);
  if (wid >= MT * NTt) return;
  int mt = wid / NTt, nt = wid % NTt;
  int lane = threadIdx.x & 31;
  int ln = lane & 15, hk = (lane >> 4) << 1;
  int row = mt * 16 + ln;
  int col = nt * 16 + ln;
  int colc = col < Nd ? col : Nd - 1;  // clamped, branch-free; garbage cols never stored
  const float* Arow = A + (size_t)row * lda + hk;
  const float* Wrow = W + (size_t)colc * Kd + hk;
  v8f acc = {};
  int ksteps = Kd >> 2;
#pragma unroll 4
  for (int ks = 0; ks < ksteps; ks++) {
    acc = wmma4(ldv2(Arow + ks * 4), ldv2(Wrow + ks * 4), acc);
  }
  int mo = (lane >> 4) * 8;
  float bsum = 0.f, bsq = 0.f;
  if (col < Nd) {
    float bv = bias ? bias[col] : 0.f;
#pragma unroll
    for (int r = 0; r < 8; r++) {
      int rr = mt * 16 + mo + r;
      float v = acc[r] + bv;
      Out[(size_t)rr * ldo + col] = v;
      bsum += v;
      bsq += v * v;
    }
  }
  if (stats) {
    bsum += __shfl_xor(bsum, 16, 32);
    bsq += __shfl_xor(bsq, 16, 32);
    if (lane < 16 && col < Nd) {
      atomicAdd(&stats[col], bsum);
      atomicAdd(&stats[Nd + col], bsq);
    }
  }
}

// -------------------------------------------- stats -> (scale, shift) pair
__global__ void finalize_bn_kernel(const float* __restrict__ stats,
                                   const float* __restrict__ g,
                                   const float* __restrict__ be, float inv_cnt,
                                   float* __restrict__ scsh, int Cn) {
  int t = blockIdx.x * blockDim.x + threadIdx.x;
  if (t >= Cn) return;
  float mean = stats[t] * inv_cnt;
  float var = stats[Cn + t] * inv_cnt - mean * mean;
  float sc = g[t] * rsqrtf(var + EPS_BN);
  scsh[t] = sc;
  scsh[Cn + t] = be[t] - mean * sc;
}

// -------------------- block5 epilogue: bn + lrelu + max/avg pool over N
__global__ void bn_lrelu_maxavg_kernel(const float* __restrict__ y,
                                       const float* __restrict__ scsh,
                                       float* __restrict__ gfeat, int B, int N,
                                       int E) {
  int t = blockIdx.x * blockDim.x + threadIdx.x;
  if (t >= B * E) return;
  int b = t / E, o = t % E;
  float sc = scsh[o], sh = scsh[E + o];
  float mx = NEG_BIG, sm = 0.f;
  for (int n = 0; n < N; n++) {
    float v = lrelu(y[((size_t)b * N + n) * E + o] * sc + sh);
    mx = fmaxf(mx, v);
    sm += v;
  }
  gfeat[(size_t)b * 2 * E + o] = mx;
  gfeat[(size_t)b * 2 * E + E + o] = sm / (float)N;
}

// ----------------------------------------- FC batch-norm + relu (in place)
__global__ void fc_bn_relu_kernel(float* __restrict__ h,
                                  const float* __restrict__ g,
                                  const float* __restrict__ be, int Rows,
                                  int Cn) {
  int t = blockIdx.x * blockDim.x + threadIdx.x;
  if (t >= Cn) return;
  float s = 0.f, q = 0.f;
  for (int r = 0; r < Rows; r++) {
    float v = h[(size_t)r * Cn + t];
    s += v;
    q += v * v;
  }
  float mean = s / (float)Rows;
  float var = q / (float)Rows - mean * mean;
  float sc = g[t] * rsqrtf(var + EPS_BN);
  float sh = be[t] - mean * sc;
  for (int r = 0; r < Rows; r++) {
    float v = h[(size_t)r * Cn + t] * sc + sh;
    h[(size_t)r * Cn + t] = v > 0.f ? v : 0.f;
  }
}

// ------------------------------------------------ log-softmax (in place)
__global__ void log_softmax_kernel(float* __restrict__ x, int Cn) {
  __shared__ float red[256];
  float* row = x + (size_t)blockIdx.x * Cn;
  int t = threadIdx.x;
  float mx = NEG_BIG;
  for (int i = t; i < Cn; i += blockDim.x) mx = fmaxf(mx, row[i]);
  red[t] = mx;
  __syncthreads();
  for (int s = 128; s > 0; s >>= 1) {
    if (t < s) red[t] = fmaxf(red[t], red[t + s]);
    __syncthreads();
  }
  mx = red[0];
  __syncthreads();
  float sm = 0.f;
  for (int i = t; i < Cn; i += blockDim.x) sm += __expf(row[i] - mx);
  red[t] = sm;
  __syncthreads();
  for (int s = 128; s > 0; s >>= 1) {
    if (t < s) red[t] += red[t + s];
    __syncthreads();
  }
  float ls = __logf(red[0]);
  __syncthreads();
  for (int i = t; i < Cn; i += blockDim.x) row[i] = row[i] - mx - ls;
}

// =========================================================== host driver
static void run_edge_block(const float* X, int stride, const float* W,
                           const float* g, const float* be, float* outFeat,
                           int ostride, int C, int O, float* nd, float* xx,
                           int* knn, float* stats, float* scsh, float* wmod,
                           hipStream_t stream) {
  const int B = 16, N = 1024;
  const int BN = B * N;
  repack_w_kernel<<<(O * C + 255) / 256, 256, 0, stream>>>(W, wmod, C, O);
  rownorm_kernel<<<(BN + 255) / 256, 256, 0, stream>>>(X, stride, C, xx, BN);
  int NT = N / 16;
  int waves = B * NT * NT;
  negdist_kernel<<<(waves + 7) / 8, 256, 0, stream>>>(X, stride, C, xx, nd, B,
                                                      N);
  topk_kernel<<<(BN + 127) / 128, 128, 0, stream>>>(nd, knn, BN, N);
  zero_kernel<<<(2 * O + 255) / 256, 256, 0, stream>>>(stats, 2 * O);
  size_t smem = 128 + (size_t)(16 * 2 * C) * sizeof(float);
  edge_pass_kernel<<<BN, 256, smem, stream>>>(X, stride, knn, wmod, C, O,
                                              stats, nullptr, nullptr, 0, 1,
                                              N);
  finalize_bn_kernel<<<(O + 255) / 256, 256, 0, stream>>>(
      stats, g, be, 1.f / ((float)BN * (float)KNN_K), scsh, O);
  edge_pass_kernel<<<BN, 256, smem, stream>>>(X, stride, knn, wmod, C, O,
                                              stats, scsh, outFeat, ostride, 2,
                                              N);
}

extern "C" void kernel_launch(void* const* d_in, const int* in_sizes, int n_in,
                              void* d_out, int out_size, void* d_ws,
                              size_t ws_size, hipStream_t stream) {
  const int B = 16, N = 1024, EMB = 1024, CLS = 751;
  const int BN = B * N;

  const float* pts = (const float*)d_in[0];
  const float* w1 = (const float*)d_in[1];
  const float* g1 = (const float*)d_in[2];
  const float* b1 = (const float*)d_in[3];
  const float* w2 = (const float*)d_in[4];
  const float* g2 = (const float*)d_in[5];
  const float* b2 = (const float*)d_in[6];
  const float* w3 = (const float*)d_in[7];
  const float* g3 = (const float*)d_in[8];
  const float* b3 = (const float*)d_in[9];
  const float* w4 = (const float*)d_in[10];
  const float* g4 = (const float*)d_in[11];
  const float* b4 = (const float*)d_in[12];
  const float* w5 = (const float*)d_in[13];
  const float* g5 = (const float*)d_in[14];
  const float* b5 = (const float*)d_in[15];
  const float* fw1 = (const float*)d_in[16];
  const float* fb1 = (const float*)d_in[17];
  const float* fg1 = (const float*)d_in[18];
  const float* fbb1 = (const float*)d_in[19];
  const float* fw2 = (const float*)d_in[20];
  const float* fb2 = (const float*)d_in[21];
  const float* fg2 = (const float*)d_in[22];
  const float* fbb2 = (const float*)d_in[23];
  const float* fw3 = (const float*)d_in[24];
  const float* fb3 = (const float*)d_in[25];
  float* out = (float*)d_out;

  // workspace carve-up (floats)
  float* fws = (float*)d_ws;
  float* nd = fws;    fws += (size_t)BN * N;          // 64 MB
  float* y5 = fws;    fws += (size_t)BN * EMB;        // 64 MB
  float* feat = fws;  fws += (size_t)BN * 512;        // 32 MB
  float* pts4 = fws;  fws += (size_t)BN * 4;
  float* w1p = fws;   fws += 64 * 8;
  float* wmod = fws;  fws += 256 * 512;               // repacked edge weights
  float* xx = fws;    fws += BN;
  float* stats = fws; fws += 2 * EMB;
  float* scsh = fws;  fws += 2 * EMB;
  float* gfeat = fws; fws += B * 2 * EMB;
  float* h1 = fws;    fws += B * 256;
  float* h2 = fws;    fws += B * 256;
  int* knn = (int*)fws;  // BN * 20 ints

  pad_pts_kernel<<<(BN + 255) / 256, 256, 0, stream>>>(pts, pts4, BN);
  pad_w1_kernel<<<1, 64, 0, stream>>>(w1, w1p);

  // 4 edge-conv blocks; features concatenated in feat[., 512]:
  //  x1 @ cols 0..63, x2 @ 64..127, x3 @ 128..255, x4 @ 256..511
  run_edge_block(pts4, 4, w1p, g1, b1, feat + 0, 512, 4, 64, nd, xx, knn,
                 stats, scsh, wmod, stream);
  run_edge_block(feat + 0, 512, w2, g2, b2, feat + 64, 512, 64, 64, nd, xx,
                 knn, stats, scsh, wmod, stream);
  run_edge_block(feat + 64, 512, w3, g3, b3, feat + 128, 512, 64, 128, nd, xx,
                 knn, stats, scsh, wmod, stream);
  run_edge_block(feat + 128, 512, w4, g4, b4, feat + 256, 512, 128, 256, nd,
                 xx, knn, stats, scsh, wmod, stream);

  // block5: y5 = feat @ w5^T, BN stats over (B*N), then bn+lrelu+max/avg pool
  zero_kernel<<<(2 * EMB + 255) / 256, 256, 0, stream>>>(stats, 2 * EMB);
  {
    int tiles = (BN / 16) * (EMB / 16);
    gemm_wt_kernel<<<(tiles + 7) / 8, 256, 0, stream>>>(
        feat, 512, w5, nullptr, y5, EMB, BN, 512, EMB, stats);
  }
  finalize_bn_kernel<<<(EMB + 255) / 256, 256, 0, stream>>>(
      stats, g5, b5, 1.f / (float)BN, scsh, EMB);
  bn_lrelu_maxavg_kernel<<<(B * EMB + 255) / 256, 256, 0, stream>>>(
      y5, scsh, gfeat, B, N, EMB);

  // FC head: fc1 -> bn/relu -> fc2 -> bn/relu -> fc3 -> log_softmax
  {
    int tiles = (B / 16) * (256 / 16);
    gemm_wt_kernel<<<(tiles + 7) / 8, 256, 0, stream>>>(
        gfeat, 2 * EMB, fw1, fb1, h1, 256, B, 2 * EMB, 256, nullptr);
  }
  fc_bn_relu_kernel<<<1, 256, 0, stream>>>(h1, fg1, fbb1, B, 256);
  {
    int tiles = (B / 16) * (256 / 16);
    gemm_wt_kernel<<<(tiles + 7) / 8, 256, 0, stream>>>(
        h1, 256, fw2, fb2, h2, 256, B, 256, 256, nullptr);
  }
  fc_bn_relu_kernel<<<1, 256, 0, stream>>>(h2, fg2, fbb2, B, 256);
  {
    int tiles = (B / 16) * ((CLS + 15) / 16);
    gemm_wt_kernel<<<(tiles + 7) / 8, 256, 0, stream>>>(
        h2, 256, fw3, fb3, out, CLS, B, 256, CLS, nullptr);
  }
  log_softmax_kernel<<<B, 256, 0, stream>>>(out, CLS);
}